// FilteredNoise_5669356835579
// MI455X (gfx1250) — compile-verified
//
#include <hip/hip_runtime.h>

// ---------------------------------------------------------------------------
// FilteredNoise (time-domain reformulation of the FFT reference)
//   B=32, FR=4000, L=65, F(hop)=64, N=2L-1=129, T=F+2L-2=192
//   ir  = (129x65 windowed-cosine matrix) @ H-row      -> WMMA f32 16x16x4
//         cosine matrix built ONCE per block in LDS, zero-padded to 144x68
//   y   = conv(noise[64], ir[129]) * 0.01  (len 192)   -> VALU from LDS
//   out = overlap-add, hop 64                          -> global_atomic_add_f32
// LDS (56.6 KB, aliased):
//   region A [0      , 39168): phase1 Mt[144][68]  -> phase2 ir[64][132]
//   region X [39168  , 56576): phase1 Hls[64][68]  -> phase2 ns[64][65]
// ---------------------------------------------------------------------------

typedef __attribute__((ext_vector_type(2))) float v2f;
typedef __attribute__((ext_vector_type(8))) float v8f;

#define FN_B        32
#define FN_FR       4000
#define FN_L        65
#define FN_HOP      64
#define FN_N        129          // IR length (odd)
#define FN_T        192          // per-frame filtered length
#define FN_OUTLEN   256128       // (FR-1)*HOP + T
#define FN_GAIN     0.01f
#define FRAMES_PER_BLOCK 64      // 4 waves x 16 frames
#define NTILES      9            // ceil(129/16) -> 144 padded cols
#define KPAD        68           // ceil(65/4)*4, zero-padded K
#define KCHUNKS     17           // K steps of 4 (f32 WMMA K=4)

#define MT_FLOATS   (144 * KPAD)             // 9792
#define IR_STRIDE   132
#define REGX_OFF    MT_FLOATS                // floats
#define SMEM_FLOATS (MT_FLOATS + FRAMES_PER_BLOCK * KPAD)   // 14144 = 56576 B

__global__ __launch_bounds__(128) void filtered_noise_fused(
    const float* __restrict__ H,      // (B*FR, 65)
    const float* __restrict__ noise,  // (B*FR, 64)
    float* __restrict__ out)          // (B, 256128), pre-zeroed
{
    __shared__ float smem[SMEM_FLOATS];
    float* Mt  = smem;               // [144][KPAD]  windowed-cosine matrix (phase 1)
    float* irs = smem;               // [64][132]    ir tile (phase 2, aliases Mt)
    float* Hls = smem + REGX_OFF;    // [64][KPAD]   zero-padded H rows (phase 1)
    float* nss = smem + REGX_OFF;    // [64][65]     noise rows (phase 2, aliases Hls)

    const int tid  = threadIdx.x;
    const int wave = tid >> 5;
    const int lane = tid & 31;
    const int mrow  = lane & 15;     // A row within tile / B column
    const int khalf = lane >> 4;     // selects K pair {0,1} vs {2,3}
    const int blockFrame = blockIdx.x * FRAMES_PER_BLOCK;

    const float TWO_PI = 6.28318530717958647692f;
    const float INV_N  = 1.0f / (float)FN_N;

    // ---- Phase 0: build cosine matrix + staged H (cooperative, once) ----
    // Mt[n][k] = hann(n)/129 * (k==0 ? 1 : 2) * cos(2*pi*k*(n-64)/129), else 0
    for (int i = tid; i < MT_FLOATS; i += 128) {
        const int n = i / KPAD;
        const int k = i - n * KPAD;
        float v = 0.0f;
        if (n < FN_N && k < FN_L) {
            const float wn = 0.5f * (1.0f - __cosf(TWO_PI * (float)n * INV_N)) * INV_N;
            const float ph = TWO_PI * (float)(n - (FN_L - 1)) * INV_N;
            const float ck = (k == 0) ? 1.0f : 2.0f;
            v = wn * ck * __cosf(ph * (float)k);
        }
        Mt[i] = v;
    }
    for (int i = tid; i < FRAMES_PER_BLOCK * KPAD; i += 128) {
        const int f = i / KPAD;
        const int k = i - f * KPAD;
        Hls[i] = (k < FN_L) ? H[(size_t)(blockFrame + f) * FN_L + k] : 0.0f;
    }
    __syncthreads();

    // ---- Phase 1: ir tiles via V_WMMA_F32_16X16X4_F32 (uniform, EXEC all-1s) ----
    // A (16x4 f32): lanes 0-15 -> M=lane, VGPR{0,1}=K{0,1}; lanes 16-31 -> K{2,3}
    // B (4x16 f32): mirrored -> lane holds col N=lane&15, K pair by lane>>4
    v8f acc[NTILES];
#pragma unroll
    for (int nt = 0; nt < NTILES; ++nt)
        acc[nt] = (v8f){0.f, 0.f, 0.f, 0.f, 0.f, 0.f, 0.f, 0.f};

    const float* aRow = &Hls[(wave * 16 + mrow) * KPAD];
    for (int kc = 0; kc < KCHUNKS; ++kc) {
        const int k0 = kc * 4 + khalf * 2;                      // even -> b64 aligned
        const v2f a = *(const v2f*)(aRow + k0);                 // ds_load_b64
#pragma unroll
        for (int nt = 0; nt < NTILES; ++nt) {
            const v2f b = *(const v2f*)&Mt[(nt * 16 + mrow) * KPAD + k0]; // ds_load_b64
            acc[nt] = __builtin_amdgcn_wmma_f32_16x16x4_f32(
                false, a, false, b, (short)0, acc[nt], false, false);
        }
    }

    __syncthreads();   // all waves done READING Mt/Hls before aliased overwrite

    // D (16x16 f32) layout: lane l, VGPR r -> M = r + 8*(l>>4), N = l&15
#pragma unroll
    for (int nt = 0; nt < NTILES; ++nt) {
        const int n = nt * 16 + mrow;
        if (n < FN_N) {
#pragma unroll
            for (int r = 0; r < 8; ++r)
                irs[(wave * 16 + r + 8 * khalf) * IR_STRIDE + n] = acc[nt][r];
        }
    }
    // stage noise into region X (Hls dead now)
    for (int i = tid; i < FRAMES_PER_BLOCK * FN_HOP; i += 128) {
        const int f = i >> 6;
        const int j = i & 63;
        nss[f * 65 + j] = noise[(size_t)(blockFrame + f) * FN_HOP + j];
    }
    __syncthreads();

    // ---- Phase 2: per-frame conv + overlap-add ----
    // filtered[f][t] = sum_j noise[f][j] * ir[f][t-j],  j in [max(0,t-128), min(63,t)]
    const int total = FRAMES_PER_BLOCK * FN_T;      // 12288 outputs
    for (int idx = tid; idx < total; idx += 128) {
        const int f = idx / FN_T;
        const int t = idx - f * FN_T;
        const int jlo = (t > FN_N - 1) ? (t - (FN_N - 1)) : 0;
        const int jhi = (t < FN_HOP - 1) ? t : (FN_HOP - 1);
        float s = 0.0f;
        const float* nf = &nss[f * 65];
        const float* rf = &irs[f * IR_STRIDE + t];
        for (int j = jlo; j <= jhi; ++j)
            s = __builtin_fmaf(nf[j], rf[-j], s);

        const int g  = blockFrame + f;              // global frame row
        const int b  = g / FN_FR;
        const int fr = g - b * FN_FR;
        atomicAdd(out + (size_t)b * FN_OUTLEN + (size_t)fr * FN_HOP + t, s * FN_GAIN);
    }
}

extern "C" void kernel_launch(void* const* d_in, const int* in_sizes, int n_in,
                              void* d_out, int out_size, void* d_ws, size_t ws_size,
                              hipStream_t stream) {
    (void)in_sizes; (void)n_in; (void)d_ws; (void)ws_size;
    const float* H     = (const float*)d_in[0];   // (32, 4000, 65)
    const float* noise = (const float*)d_in[1];   // (32, 4000, 64)
    float* out = (float*)d_out;                   // (32, 256128)

    // Overlap-add accumulates with atomics: output must start at zero.
    hipMemsetAsync(d_out, 0, (size_t)out_size * sizeof(float), stream);

    const int totalFrames = FN_B * FN_FR;                       // 128000
    const int blocks = totalFrames / FRAMES_PER_BLOCK;          // 2000
    filtered_noise_fused<<<blocks, 128, 0, stream>>>(H, noise, out);
}